// OHEMloss_55473797595138
// MI455X (gfx1250) — compile-verified
//
#include <hip/hip_runtime.h>
#include <hip/hip_bf16.h>

// ---------------------------------------------------------------------------
// OHEM (hard-negative-mining) MSE loss, CRAFT-style, via exact radix-select.
// 3 streaming sweeps (HBM once, then L2-resident) + tiny select kernels.
// Pass 1 streams through LDS with gfx1250 async global->LDS copies
// (GLOBAL_LOAD_ASYNC_TO_LDS_B128, ASYNCcnt double-buffering).
// Radix digits over float bit pattern (loss >= 0 so uint order == float order):
//   level1: bits[31:21] (11b, 2048 bins, count + f64 sum)
//   level2: bits[20:10] (11b, 2048 bins, count + f64 sum)
//   level3: bits[ 9: 0] (10b, 1024 bins, count only; bin == exact value)
// ---------------------------------------------------------------------------

#define L12_BINS 2048
#define L3_BINS  1024

// per-channel workspace layout (byte offsets)
#define CNT1_OFF   0        // 2048 u32
#define CNT2_OFF   8192     // 2048 u32
#define CNT3_OFF   16384    // 1024 u32
#define STATSU_OFF 20480    // 16 u32: [0]=num_pos [1]=num_neg [2]=k [3]=b1 [4]=r1 [5]=b2 [6]=r2
#define SUM1_OFF   20544    // 2048 f64
#define SUM2_OFF   36928    // 2048 f64
#define STATSD_OFF 53312    // 8 f64: [0]=pos_sum [1]=sum_above1 [2]=sum_above2
#define CH_STRIDE  53376    // bytes per channel (8-aligned)

#define POS_T 0.1f
#define NEG_T 0.0f
#define SENTINEL 0xFFFFFFFFu

// ---------------------------------------------------------------------------
// gfx1250 async global->LDS copy (tracked by ASYNCcnt).
// the LDS address is the LDS base plus the VDST value; low 32 bits of a flat LDS pointer are the
// LDS byte offset (flat-aperture semantics), which is what VDST must hold.
__device__ __forceinline__ void async_ld_b128(void* lds_ptr, const void* gptr) {
    unsigned lds_off = (unsigned)(size_t)lds_ptr;
    asm volatile("global_load_async_to_lds_b128 %0, %1, off"
                 :: "v"(lds_off), "v"(gptr)
                 : "memory");
}

#if defined(__has_builtin)
#if __has_builtin(__builtin_amdgcn_s_wait_asynccnt)
#define HAVE_WAIT_ASYNC_BUILTIN 1
#endif
#endif

__device__ __forceinline__ void wait_asynccnt_4() {
    asm volatile("s_wait_asynccnt 0x4" ::: "memory");
}
__device__ __forceinline__ void wait_asynccnt_0() {
#ifdef HAVE_WAIT_ASYNC_BUILTIN
    __builtin_amdgcn_s_wait_asynccnt(0);
    asm volatile("" ::: "memory");
#else
    asm volatile("s_wait_asynccnt 0x0" ::: "memory");
#endif
}

// ---------------------------------------------------------------------------
__global__ void zero_ws_kernel(unsigned* __restrict__ ws, int nwords) {
    int stride = gridDim.x * blockDim.x;
    for (int i = blockIdx.x * blockDim.x + threadIdx.x; i < nwords; i += stride)
        ws[i] = 0u;
}

// ---------------------------------------------------------------------------
__device__ __forceinline__ void acc_elem(float p, float t,
                                         unsigned* hcnt, double* hsum,
                                         unsigned* pcnt, double* psum,
                                         unsigned* ncnt) {
    float d = p - t;
    float l = d * d;
    if (t >= POS_T) {
        atomicAdd(pcnt, 1u);
        unsafeAtomicAdd(psum, (double)l);
    } else if (t <= NEG_T) {
        atomicAdd(ncnt, 1u);
        unsigned key = __float_as_uint(l) >> 21;     // 11-bit level-1 digit
        atomicAdd(hcnt + key, 1u);
        unsafeAtomicAdd(hsum + key, (double)l);
    }
}

// ---------------------------------------------------------------------------
// Pass 1: the HBM-bound sweep. Double-buffered async global->LDS staging:
// each thread async-loads the 64 bytes it will itself consume next tile,
// so per-wave ASYNCcnt ordering is the only synchronization needed.
#define P1_THREADS 512

__global__ __launch_bounds__(P1_THREADS)
void pass1_hist(const float* __restrict__ out2, const float* __restrict__ cmap,
                const float* __restrict__ amap, unsigned char* __restrict__ ws,
                int n4) {
    __shared__ unsigned s_cnt[2][L12_BINS];
    __shared__ double   s_sum[2][L12_BINS];
    __shared__ unsigned s_pos[2], s_neg[2];
    __shared__ double   s_psum[2];
    __shared__ float4   st_out[2][P1_THREADS * 2];   // 32KB
    __shared__ float4   st_c[2][P1_THREADS];         // 16KB
    __shared__ float4   st_a[2][P1_THREADS];         // 16KB

    for (int i = threadIdx.x; i < L12_BINS; i += blockDim.x) {
        s_cnt[0][i] = 0u; s_cnt[1][i] = 0u;
        s_sum[0][i] = 0.0; s_sum[1][i] = 0.0;
    }
    if (threadIdx.x < 2) {
        s_pos[threadIdx.x] = 0u; s_neg[threadIdx.x] = 0u; s_psum[threadIdx.x] = 0.0;
    }
    __syncthreads();

    const float4* o4 = (const float4*)out2;
    const float4* c4 = (const float4*)cmap;
    const float4* a4 = (const float4*)amap;
    int  tid = threadIdx.x;
    long stride = (long)gridDim.x * P1_THREADS;
    long v = (long)blockIdx.x * P1_THREADS + tid;

    // prologue: stage first tile into buffer 0
    if (v < n4) {
        async_ld_b128(&st_out[0][2 * tid],     &o4[2 * v]);
        async_ld_b128(&st_out[0][2 * tid + 1], &o4[2 * v + 1]);
        async_ld_b128(&st_c[0][tid],           &c4[v]);
        async_ld_b128(&st_a[0][tid],           &a4[v]);
    }

    int buf = 0;
    for (; v < n4; v += stride) {
        long vn  = v + stride;
        int nbuf = buf ^ 1;
        if (vn < n4) {
            // stage next tile, then wait for the 4 older loads (in-order done)
            async_ld_b128(&st_out[nbuf][2 * tid],     &o4[2 * vn]);
            async_ld_b128(&st_out[nbuf][2 * tid + 1], &o4[2 * vn + 1]);
            async_ld_b128(&st_c[nbuf][tid],           &c4[vn]);
            async_ld_b128(&st_a[nbuf][tid],           &a4[vn]);
            wait_asynccnt_4();
        } else {
            wait_asynccnt_0();
        }

        float4 tc = st_c[buf][tid];
        float4 ta = st_a[buf][tid];
        float4 oA = st_out[buf][2 * tid];       // {e0c0,e0c1,e1c0,e1c1}
        float4 oB = st_out[buf][2 * tid + 1];   // {e2c0,e2c1,e3c0,e3c1}
        // channel 0 (character)
        acc_elem(oA.x, tc.x, s_cnt[0], s_sum[0], &s_pos[0], &s_psum[0], &s_neg[0]);
        acc_elem(oA.z, tc.y, s_cnt[0], s_sum[0], &s_pos[0], &s_psum[0], &s_neg[0]);
        acc_elem(oB.x, tc.z, s_cnt[0], s_sum[0], &s_pos[0], &s_psum[0], &s_neg[0]);
        acc_elem(oB.z, tc.w, s_cnt[0], s_sum[0], &s_pos[0], &s_psum[0], &s_neg[0]);
        // channel 1 (affinity)
        acc_elem(oA.y, ta.x, s_cnt[1], s_sum[1], &s_pos[1], &s_psum[1], &s_neg[1]);
        acc_elem(oA.w, ta.y, s_cnt[1], s_sum[1], &s_pos[1], &s_psum[1], &s_neg[1]);
        acc_elem(oB.y, ta.z, s_cnt[1], s_sum[1], &s_pos[1], &s_psum[1], &s_neg[1]);
        acc_elem(oB.w, ta.w, s_cnt[1], s_sum[1], &s_pos[1], &s_psum[1], &s_neg[1]);
        buf = nbuf;
    }
    __syncthreads();

    for (int ch = 0; ch < 2; ++ch) {
        unsigned char* base = ws + (size_t)ch * CH_STRIDE;
        unsigned* g_cnt = (unsigned*)(base + CNT1_OFF);
        double*   g_sum = (double*)(base + SUM1_OFF);
        for (int b = threadIdx.x; b < L12_BINS; b += blockDim.x) {
            unsigned c = s_cnt[ch][b];
            if (c) {
                atomicAdd(&g_cnt[b], c);
                unsafeAtomicAdd(&g_sum[b], s_sum[ch][b]);
            }
        }
    }
    if (threadIdx.x == 0) {
        for (int ch = 0; ch < 2; ++ch) {
            unsigned char* base = ws + (size_t)ch * CH_STRIDE;
            unsigned* su = (unsigned*)(base + STATSU_OFF);
            double*   sd = (double*)(base + STATSD_OFF);
            atomicAdd(&su[0], s_pos[ch]);
            atomicAdd(&su[1], s_neg[ch]);
            unsafeAtomicAdd(&sd[0], s_psum[ch]);
        }
    }
}

// ---------------------------------------------------------------------------
// One block per channel. Find boundary bin for top-k, sum of bins above it.
__global__ __launch_bounds__(256)
void select_kernel(unsigned char* __restrict__ ws, int level) {
    int ch = blockIdx.x;
    unsigned char* base = ws + (size_t)ch * CH_STRIDE;
    unsigned* su = (unsigned*)(base + STATSU_OFF);
    double*   sd = (double*)(base + STATSD_OFF);
    const unsigned* cnt = (const unsigned*)(base + (level == 1 ? CNT1_OFF : CNT2_OFF));
    const double*   sum = (const double*)(base + (level == 1 ? SUM1_OFF : SUM2_OFF));

    __shared__ unsigned s_cnt[L12_BINS];
    __shared__ unsigned s_scan[256];
    __shared__ unsigned s_k, s_b, s_r;
    __shared__ double   s_above;

    for (int i = threadIdx.x; i < L12_BINS; i += 256) s_cnt[i] = cnt[i];
    if (threadIdx.x == 0) {
        unsigned k;
        if (level == 1) {
            unsigned np = su[0], nn = su[1];
            unsigned long long kk = 3ull * (unsigned long long)np;
            if (kk < 1000ull) kk = 1000ull;
            if (kk > (unsigned long long)nn) kk = nn;
            k = (unsigned)kk;
            su[2] = k;
        } else {
            k = su[4];   // r1 from level 1
        }
        s_k = k;
        s_b = SENTINEL;
        s_r = 0u;
        s_above = 0.0;
    }
    __syncthreads();
    unsigned k = s_k;

    // each thread owns 8 bins in descending order; group g covers
    // bins [L12_BINS - 8*(g+1), L12_BINS - 8*g)
    const int GB = L12_BINS / 256;   // 8
    int hi = L12_BINS - GB * threadIdx.x;   // exclusive upper bin of my group
    unsigned gsum = 0;
    for (int j = 1; j <= GB; ++j) gsum += s_cnt[hi - j];

    s_scan[threadIdx.x] = gsum;
    __syncthreads();
    for (int off = 1; off < 256; off <<= 1) {
        unsigned add = (threadIdx.x >= (unsigned)off) ? s_scan[threadIdx.x - off] : 0u;
        __syncthreads();
        s_scan[threadIdx.x] += add;
        __syncthreads();
    }
    unsigned cum = s_scan[threadIdx.x];
    unsigned before = cum - gsum;

    if (k > 0 && before < k && k <= cum) {   // exactly one thread crosses
        unsigned c = before;
        for (int j = 1; j <= GB; ++j) {
            int bin = hi - j;
            unsigned cc = s_cnt[bin];
            if (c + cc >= k) { s_b = (unsigned)bin; s_r = k - c; break; }
            c += cc;
        }
    }
    __syncthreads();
    unsigned b = s_b;

    double local = 0.0;
    if (b != SENTINEL) {
        for (int i = threadIdx.x; i < L12_BINS; i += 256)
            if ((unsigned)i > b) local += sum[i];
    }
    unsafeAtomicAdd(&s_above, local);
    __syncthreads();

    if (threadIdx.x == 0) {
        if (level == 1) { su[3] = b; su[4] = s_r; sd[1] = s_above; }
        else           { su[5] = b; su[6] = s_r; sd[2] = s_above; }
    }
}

// ---------------------------------------------------------------------------
__global__ __launch_bounds__(1024)
void pass2_hist(const float* __restrict__ out2, const float* __restrict__ cmap,
                const float* __restrict__ amap, unsigned char* __restrict__ ws,
                int n4) {
    __shared__ unsigned s_cnt[2][L12_BINS];
    __shared__ double   s_sum[2][L12_BINS];
    for (int i = threadIdx.x; i < L12_BINS; i += blockDim.x) {
        s_cnt[0][i] = 0u; s_cnt[1][i] = 0u;
        s_sum[0][i] = 0.0; s_sum[1][i] = 0.0;
    }
    unsigned b1c = ((const unsigned*)(ws + STATSU_OFF))[3];
    unsigned b1a = ((const unsigned*)(ws + CH_STRIDE + STATSU_OFF))[3];
    __syncthreads();

    const float4* o4 = (const float4*)out2;
    const float4* c4 = (const float4*)cmap;
    const float4* a4 = (const float4*)amap;
    long stride = (long)gridDim.x * blockDim.x;
    for (long v = (long)blockIdx.x * blockDim.x + threadIdx.x; v < n4; v += stride) {
        __builtin_prefetch(&c4[v + stride]);
        __builtin_prefetch(&a4[v + stride]);
        __builtin_prefetch(&o4[2 * (v + stride)]);
        float4 tc = c4[v];
        float4 ta = a4[v];
        float4 oA = o4[2 * v];
        float4 oB = o4[2 * v + 1];
        float p[2][4] = {{oA.x, oA.z, oB.x, oB.z}, {oA.y, oA.w, oB.y, oB.w}};
        float t[2][4] = {{tc.x, tc.y, tc.z, tc.w}, {ta.x, ta.y, ta.z, ta.w}};
        unsigned b1v[2] = {b1c, b1a};
        #pragma unroll
        for (int ch = 0; ch < 2; ++ch) {
            #pragma unroll
            for (int e = 0; e < 4; ++e) {
                float tt = t[ch][e];
                if (tt <= NEG_T) {
                    float d = p[ch][e] - tt;
                    float l = d * d;
                    unsigned bits = __float_as_uint(l);
                    if ((bits >> 21) == b1v[ch]) {
                        unsigned key = (bits >> 10) & 0x7FFu;
                        atomicAdd(&s_cnt[ch][key], 1u);
                        unsafeAtomicAdd(&s_sum[ch][key], (double)l);
                    }
                }
            }
        }
    }
    __syncthreads();

    for (int ch = 0; ch < 2; ++ch) {
        unsigned char* base = ws + (size_t)ch * CH_STRIDE;
        unsigned* g_cnt = (unsigned*)(base + CNT2_OFF);
        double*   g_sum = (double*)(base + SUM2_OFF);
        for (int b = threadIdx.x; b < L12_BINS; b += blockDim.x) {
            unsigned c = s_cnt[ch][b];
            if (c) {
                atomicAdd(&g_cnt[b], c);
                unsafeAtomicAdd(&g_sum[b], s_sum[ch][b]);
            }
        }
    }
}

// ---------------------------------------------------------------------------
__global__ __launch_bounds__(1024)
void pass3_hist(const float* __restrict__ out2, const float* __restrict__ cmap,
                const float* __restrict__ amap, unsigned char* __restrict__ ws,
                int n4) {
    __shared__ unsigned s_cnt[2][L3_BINS];
    for (int i = threadIdx.x; i < L3_BINS; i += blockDim.x) {
        s_cnt[0][i] = 0u; s_cnt[1][i] = 0u;
    }
    const unsigned* suc = (const unsigned*)(ws + STATSU_OFF);
    const unsigned* sua = (const unsigned*)(ws + CH_STRIDE + STATSU_OFF);
    unsigned pref[2];   // (b1<<11)|b2, compared against bits[31:10]
    pref[0] = (suc[3] << 11) | suc[5];
    pref[1] = (sua[3] << 11) | sua[5];
    __syncthreads();

    const float4* o4 = (const float4*)out2;
    const float4* c4 = (const float4*)cmap;
    const float4* a4 = (const float4*)amap;
    long stride = (long)gridDim.x * blockDim.x;
    for (long v = (long)blockIdx.x * blockDim.x + threadIdx.x; v < n4; v += stride) {
        __builtin_prefetch(&c4[v + stride]);
        __builtin_prefetch(&a4[v + stride]);
        __builtin_prefetch(&o4[2 * (v + stride)]);
        float4 tc = c4[v];
        float4 ta = a4[v];
        float4 oA = o4[2 * v];
        float4 oB = o4[2 * v + 1];
        float p[2][4] = {{oA.x, oA.z, oB.x, oB.z}, {oA.y, oA.w, oB.y, oB.w}};
        float t[2][4] = {{tc.x, tc.y, tc.z, tc.w}, {ta.x, ta.y, ta.z, ta.w}};
        #pragma unroll
        for (int ch = 0; ch < 2; ++ch) {
            #pragma unroll
            for (int e = 0; e < 4; ++e) {
                float tt = t[ch][e];
                if (tt <= NEG_T) {
                    float d = p[ch][e] - tt;
                    float l = d * d;
                    unsigned bits = __float_as_uint(l);
                    if ((bits >> 10) == pref[ch])
                        atomicAdd(&s_cnt[ch][bits & 0x3FFu], 1u);
                }
            }
        }
    }
    __syncthreads();

    for (int ch = 0; ch < 2; ++ch) {
        unsigned* g_cnt = (unsigned*)(ws + (size_t)ch * CH_STRIDE + CNT3_OFF);
        for (int b = threadIdx.x; b < L3_BINS; b += blockDim.x) {
            unsigned c = s_cnt[ch][b];
            if (c) atomicAdd(&g_cnt[b], c);
        }
    }
}

// ---------------------------------------------------------------------------
__global__ __launch_bounds__(256)
void final_kernel(unsigned char* __restrict__ ws, float* __restrict__ out) {
    __shared__ unsigned s_cnt3[2][L3_BINS];
    __shared__ double s_loss[2];
    for (int ch = 0; ch < 2; ++ch) {
        const unsigned* g = (const unsigned*)(ws + (size_t)ch * CH_STRIDE + CNT3_OFF);
        for (int i = threadIdx.x; i < L3_BINS; i += 256) s_cnt3[ch][i] = g[i];
    }
    __syncthreads();

    if (threadIdx.x < 2) {
        int ch = threadIdx.x;
        const unsigned* su = (const unsigned*)(ws + (size_t)ch * CH_STRIDE + STATSU_OFF);
        const double*   sd = (const double*)(ws + (size_t)ch * CH_STRIDE + STATSD_OFF);
        unsigned np = su[0], k = su[2], b1 = su[3], b2 = su[5], r2 = su[6];
        double sum_top = 0.0;
        unsigned cum = 0;
        if (r2 > 0 && b1 != SENTINEL && b2 != SENTINEL) {
            for (int b = L3_BINS - 1; b >= 0; --b) {
                unsigned c = s_cnt3[ch][b];
                if (!c) continue;
                float v = __uint_as_float((b1 << 21) | (b2 << 10) | (unsigned)b);
                unsigned take = (cum + c <= r2) ? c : (r2 - cum);
                sum_top += (double)take * (double)v;
                cum += take;
                if (cum >= r2) break;
            }
        }
        double neg_sum = sd[1] + sd[2] + sum_top;
        double denom = (double)np + (double)k;
        s_loss[ch] = (denom > 0.0) ? (sd[0] + neg_sum) / denom : 0.0;
    }
    __syncthreads();
    if (threadIdx.x == 0) out[0] = (float)(2.0 * s_loss[0] + s_loss[1]);
}

// ---------------------------------------------------------------------------
extern "C" void kernel_launch(void* const* d_in, const int* in_sizes, int n_in,
                              void* d_out, int out_size, void* d_ws, size_t ws_size,
                              hipStream_t stream) {
    const float* out2 = (const float*)d_in[0];   // (B,H,W,2)
    const float* cmap = (const float*)d_in[1];   // (B,H,W)
    const float* amap = (const float*)d_in[2];   // (B,H,W)
    unsigned char* ws = (unsigned char*)d_ws;

    int N  = in_sizes[1];        // B*H*W
    int n4 = N / 4;              // elements processed 4-wide (N divisible by 4)
    int nwords = (2 * CH_STRIDE) / 4;

    zero_ws_kernel<<<64, 256, 0, stream>>>((unsigned*)ws, nwords);
    pass1_hist   <<<512, P1_THREADS, 0, stream>>>(out2, cmap, amap, ws, n4);
    select_kernel<<<2, 256, 0, stream>>>(ws, 1);
    pass2_hist   <<<512, 1024, 0, stream>>>(out2, cmap, amap, ws, n4);
    select_kernel<<<2, 256, 0, stream>>>(ws, 2);
    pass3_hist   <<<512, 1024, 0, stream>>>(out2, cmap, amap, ws, n4);
    final_kernel <<<1, 256, 0, stream>>>(ws, (float*)d_out);
}